// RankingLoss_1803886265544
// MI455X (gfx1250) — compile-verified
//
#include <hip/hip_runtime.h>
#include <hip/hip_bf16.h>

// Ranking (hinge) loss over all pairs, B=8192, N_CLASSES=4, MARGIN=0.
// Phase 1: risk = hazards.sum(axis=1) via V_WMMA_F32_16X16X4_F32 (A = 16x4
//          hazards tile, B = ones 4x16  ->  every D column holds risk[m]).
// Phase 2: O(B^2) VALU pair loop, LDS-staged j tiles, censoring folded into
//          the class key so the inner loop has no mask math.
// Phase 3: deterministic tree reductions; loss = total / count (0 if count==0).

typedef __attribute__((ext_vector_type(2))) float v2f;
typedef __attribute__((ext_vector_type(8))) float v8f;

#define BN      8192
#define GI      32              // blocks over i (256 rows each)
#define GJ      16              // blocks over j
#define JCHUNK  (BN / GJ)       // 512 j's staged per block
#define NBLK    (GI * GJ)       // 512 partial sums

// ---------- Phase 1: risk[i] = sum_k hazards[i][k] via f32 WMMA ----------
__global__ __launch_bounds__(256) void risk_wmma_kernel(
    const float* __restrict__ hazards, float* __restrict__ risk) {
    const int lane = threadIdx.x & 31;
    const int wave = threadIdx.x >> 5;
    const int tile = (blockIdx.x * 8 + wave) * 16;   // 16 rows per wave

    // A-matrix 16x4 f32 layout: lanes 0-15 hold K=0,1 ; lanes 16-31 hold K=2,3
    const int m  = lane & 15;
    const int kb = (lane >> 4) << 1;
    v2f a = *(const v2f*)(hazards + (tile + m) * 4 + kb);   // 8B aligned

    v2f b = {1.0f, 1.0f};   // ones matrix -> row sum
    v8f cacc = {};
    v8f d = __builtin_amdgcn_wmma_f32_16x16x4_f32(
        /*neg_a=*/false, a, /*neg_b=*/false, b,
        /*c_mod=*/(short)0, cacc, /*reuse_a=*/false, /*reuse_b=*/false);

    // D[m][n] = risk[tile+m] for all n.  Column 0 lives in lane 0 (m=0..7)
    // and lane 16 (m=8..15): two 32-byte stores cover the 16 risks.
    if (lane == 0 || lane == 16) {
        const int base = tile + ((lane == 16) ? 8 : 0);
        float4 lo = make_float4(d[0], d[1], d[2], d[3]);
        float4 hi = make_float4(d[4], d[5], d[6], d[7]);
        *(float4*)(risk + base)     = lo;
        *(float4*)(risk + base + 4) = hi;
    }
}

// ---------- Phase 2: pairwise hinge accumulation ----------
__global__ __launch_bounds__(256) void pair_kernel(
    const float* __restrict__ risk,
    const int*   __restrict__ Y,
    const int*   __restrict__ c,
    float* __restrict__ tsum_out,
    int*   __restrict__ tcnt_out) {
    __shared__ float2 sj[JCHUNK];
    __shared__ float  rs[256];
    __shared__ int    rc[256];

    const int tid   = threadIdx.x;
    const int i     = blockIdx.x * 256 + tid;
    const int jbase = blockIdx.y * JCHUNK;

    // Stage (risk[j], Y[j]) interleaved -> one ds_load_b64 per inner step.
    for (int t = tid; t < JCHUNK; t += 256) {
        const int jj = jbase + t;
        sj[t] = make_float2(risk[jj], (float)Y[jj]);
    }
    __syncthreads();

    const float ri = risk[i];
    // Censored rows can never satisfy Y[j] > yi_eff -> mask folded for free.
    const float yi = (c[i] == 0) ? (float)Y[i] : 3.0e38f;

    float tsum = 0.0f;
    int   tcnt = 0;
#pragma unroll 8
    for (int k = 0; k < JCHUNK; ++k) {
        const float2 p = sj[k];
        const bool pred = p.y > yi;                 // mask(i,j)
        const float dif = fmaxf(p.x - ri, 0.0f);    // relu(rj - ri), MARGIN=0
        tsum += pred ? dif : 0.0f;
        tcnt += pred ? 1 : 0;
    }

    // Deterministic block tree reduction.
    rs[tid] = tsum; rc[tid] = tcnt;
    __syncthreads();
    for (int s = 128; s > 0; s >>= 1) {
        if (tid < s) { rs[tid] += rs[tid + s]; rc[tid] += rc[tid + s]; }
        __syncthreads();
    }
    if (tid == 0) {
        const int b = blockIdx.y * gridDim.x + blockIdx.x;
        tsum_out[b] = rs[0];
        tcnt_out[b] = rc[0];
    }
}

// ---------- Phase 3: final reduction + divide ----------
__global__ __launch_bounds__(256) void finalize_kernel(
    const float* __restrict__ tsum_in,
    const int*   __restrict__ tcnt_in,
    float* __restrict__ out) {
    __shared__ float rs[256];
    __shared__ int   rc[256];
    const int tid = threadIdx.x;
    float s = 0.0f; int n = 0;
    for (int t = tid; t < NBLK; t += 256) { s += tsum_in[t]; n += tcnt_in[t]; }
    rs[tid] = s; rc[tid] = n;
    __syncthreads();
    for (int st = 128; st > 0; st >>= 1) {
        if (tid < st) { rs[tid] += rs[tid + st]; rc[tid] += rc[tid + st]; }
        __syncthreads();
    }
    if (tid == 0)
        out[0] = (rc[0] > 0) ? (rs[0] / (float)rc[0]) : 0.0f;
}

extern "C" void kernel_launch(void* const* d_in, const int* in_sizes, int n_in,
                              void* d_out, int out_size, void* d_ws, size_t ws_size,
                              hipStream_t stream) {
    (void)in_sizes; (void)n_in; (void)out_size; (void)ws_size;
    const float* hazards = (const float*)d_in[0];
    // d_in[1] = S (unused by the loss)
    const int* Y = (const int*)d_in[2];
    const int* c = (const int*)d_in[3];

    float* wsf  = (float*)d_ws;
    float* risk = wsf;                       // [0, 8192)
    float* tsum = wsf + BN;                  // [8192, 8192+512)
    int*   tcnt = (int*)(wsf + BN + NBLK);   // next 512 ints
    float* out  = (float*)d_out;

    risk_wmma_kernel<<<BN / 128, 256, 0, stream>>>(hazards, risk);

    dim3 grid(GI, GJ);
    pair_kernel<<<grid, 256, 0, stream>>>(risk, Y, c, tsum, tcnt);

    finalize_kernel<<<1, 256, 0, stream>>>(tsum, tcnt, out);
}